// MPSLayer_18356690223107
// MI455X (gfx1250) — compile-verified
//
#include <hip/hip_runtime.h>

#define N_SITES   22
#define CHI       32
#define NMID      (N_SITES - 2)      // 20 middle cores
#define NBONDS    (N_SITES - 1)      // 21 entropies
#define HALF_ROWS 2048               // 2^11
#define STATE_SIZE (1u << N_SITES)   // 4194304
#define EPSF      1e-10f

typedef __attribute__((ext_vector_type(2))) float v2f;
typedef __attribute__((ext_vector_type(8))) float v8f;

// Intra-wave LDS ordering (single-wave workgroup: s_barrier is a NOP, so wait on DScnt).
__device__ __forceinline__ void lds_sync() {
  asm volatile("s_wait_dscnt 0" ::: "memory");
}

// ---------------------------------------------------------------------------
// Gram-matrix chain: G_0 from core_first; G_{k+1} = sum_b M_b^T G_k M_b.
// One block of 1024 threads (32 waves), 21 grams written to ws.
// ---------------------------------------------------------------------------
__global__ void gram_chain_kernel(const float* __restrict__ core_first,
                                  const float* __restrict__ cores_mid,
                                  float* __restrict__ grams) {
  __shared__ float Gs[32][32];
  __shared__ float Tt[2][32][32];
  __shared__ float Gn[32][32];
  const int tid = threadIdx.x;
  const int i = tid >> 5, j = tid & 31;

  // G0[i][j] = sum_b cf[b*32+i]*cf[b*32+j]   (core_first is [1,2,32])
  Gs[i][j] = core_first[i] * core_first[j] + core_first[32 + i] * core_first[32 + j];
  __syncthreads();

  for (int k = 0; k < NBONDS; ++k) {
    grams[k * 1024 + tid] = Gs[i][j];
    if (k == NMID) break;
    const float* core = cores_mid + k * (CHI * 2 * CHI);  // [32][2][32]
    // phase 1: T[b][a=i][c=j] = sum_a' G[i][a'] * core[a'][b][j]
    float t0 = 0.f, t1 = 0.f;
    for (int a2 = 0; a2 < 32; ++a2) {
      float g = Gs[i][a2];
      t0 += g * core[a2 * 64 + j];
      t1 += g * core[a2 * 64 + 32 + j];
    }
    Tt[0][i][j] = t0;
    Tt[1][i][j] = t1;
    __syncthreads();
    // phase 2: Gn[c=i][c'=j] = sum_b sum_a core[a][b][i] * T[b][a][j]
    float acc = 0.f;
    for (int a = 0; a < 32; ++a) {
      acc += core[a * 64 + i] * Tt[0][a][j];
      acc += core[a * 64 + 32 + i] * Tt[1][a][j];
    }
    Gn[i][j] = acc;
    __syncthreads();
    Gs[i][j] = Gn[i][j];
    __syncthreads();
  }
}

// ---------------------------------------------------------------------------
// Jacobi eigensolver + von Neumann entropy. One wave32 per 32x32 gram.
// ---------------------------------------------------------------------------
__global__ void entropy_kernel(const float* __restrict__ grams,
                               float* __restrict__ out_ent) {
  __shared__ float A[32][33];
  const int t = threadIdx.x;  // 0..31
  const float* G = grams + blockIdx.x * (CHI * CHI);
  for (int j = 0; j < 32; ++j) A[t][j] = G[t * 32 + j];
  lds_sync();

#pragma unroll 1
  for (int sweep = 0; sweep < 14; ++sweep) {
#pragma unroll 1
    for (int p = 0; p < 31; ++p) {
#pragma unroll 1
      for (int q = p + 1; q < 32; ++q) {
        float apq = A[p][q];                 // uniform across lanes
        if (fabsf(apq) > 1e-36f) {
          float app = A[p][p], aqq = A[q][q];
          float theta = 0.5f * (aqq - app) / apq;
          float tt = copysignf(1.0f, theta) /
                     (fabsf(theta) + sqrtf(1.0f + theta * theta));
          float c = 1.0f / sqrtf(1.0f + tt * tt);
          float s = tt * c;
          // column rotation (A <- A J): lane t updates row t
          float aip = A[t][p], aiq = A[t][q];
          A[t][p] = c * aip - s * aiq;
          A[t][q] = s * aip + c * aiq;
          lds_sync();
          // row rotation (A <- J^T A): lane t updates column t
          float apj = A[p][t], aqj = A[q][t];
          A[p][t] = c * apj - s * aqj;
          A[q][t] = s * apj + c * aqj;
          lds_sync();
        }
      }
    }
  }

  float pv = fmaxf(A[t][t], 0.0f);
  float tot = pv;
  for (int m = 16; m >= 1; m >>= 1) tot += __shfl_xor(tot, m, 32);
  float pn = pv / tot;
  float term = (pn > EPSF) ? (-pn * logf(fmaxf(pn, EPSF))) : 0.0f;
  float ent = term;
  for (int m = 16; m >= 1; m >>= 1) ent += __shfl_xor(ent, m, 32);
  if (t == 0) out_ent[blockIdx.x] = ent;
}

// ---------------------------------------------------------------------------
// Half-chain builders.
// ---------------------------------------------------------------------------
__global__ void init_halves_kernel(const float* __restrict__ core_first,
                                   const float* __restrict__ core_last,
                                   float* __restrict__ L0, float* __restrict__ R0) {
  int t = threadIdx.x;  // 0..63
  L0[t] = core_first[t];                 // [2,32]: rows by b0
  int b = t >> 5, a = t & 31;
  R0[t] = core_last[a * 2 + b];          // R0[b21*32+a] = core_last[a][b21][0]
}

// out[(r*2+b)*32 + c] = sum_a in[r*32+a] * core[a][b][c]
__global__ void left_step_kernel(const float* __restrict__ in,
                                 const float* __restrict__ core,
                                 float* __restrict__ out, int total) {
  int tid = blockIdx.x * blockDim.x + threadIdx.x;
  if (tid >= total) return;
  int c = tid & 31;
  int rn = tid >> 5;
  int b = rn & 1;
  int r = rn >> 1;
  const float* inr = in + r * CHI;
  const float* cp = core + b * 32 + c;
  float acc = 0.f;
  for (int a = 0; a < 32; ++a) acc += inr[a] * cp[a * 64];
  out[tid] = acc;
}

// out[(b<<rbits | r)*32 + a] = sum_a' core[a][b][a'] * in[r*32 + a']
__global__ void right_step_kernel(const float* __restrict__ in,
                                  const float* __restrict__ core,
                                  float* __restrict__ out, int rbits, int total) {
  int tid = blockIdx.x * blockDim.x + threadIdx.x;
  if (tid >= total) return;
  int a = tid & 31;
  int rn = tid >> 5;
  int b = rn >> rbits;
  int r = rn & ((1 << rbits) - 1);
  const float* inr = in + r * CHI;
  const float* cp = core + a * 64 + b * 32;
  float acc = 0.f;
  for (int ap = 0; ap < 32; ++ap) acc += cp[ap] * inr[ap];
  out[tid] = acc;
}

// ---------------------------------------------------------------------------
// Final GEMM: state[rl*2048+rr] = sum_a L[rl,a]*R[rr,a] via V_WMMA_F32_16X16X4_F32.
// Block = 256 threads = 8 waves; each wave owns one 16x16 tile; block covers 32x64.
// ---------------------------------------------------------------------------
__global__ void final_gemm_kernel(const float* __restrict__ L,
                                  const float* __restrict__ R,
                                  float* __restrict__ out) {
  const int tid = threadIdx.x;
  const int lane = tid & 31;
  const int wid = tid >> 5;        // 0..7
  const int half = lane >> 4;      // 0/1
  const int idx = lane & 15;       // 0..15
  const int m0 = blockIdx.y * 32 + (wid >> 2) * 16;
  const int n0 = blockIdx.x * 64 + (wid & 3) * 16;

  const v2f* Lv = (const v2f*)L;   // row stride = 16 v2f
  const v2f* Rv = (const v2f*)R;
  const int arow = (m0 + idx) * 16;
  const int brow = (n0 + idx) * 16;

  v8f c = {};
#pragma unroll
  for (int k0 = 0; k0 < 8; ++k0) {
    // A fragment (16x4): lane = M + 16*(K/2), holds K%2 pair -> floats [4k0+2*half, +1]
    v2f a = Lv[arow + 2 * k0 + half];
    // B fragment (4x16): lane = N + 16*(K/2) -> same contiguous pair from R row
    v2f b = Rv[brow + 2 * k0 + half];
    c = __builtin_amdgcn_wmma_f32_16x16x4_f32(false, a, false, b,
                                              (short)0, c, false, false);
  }

#pragma unroll
  for (int v = 0; v < 8; ++v) {
    int m = m0 + v + 8 * half;     // C/D layout: VGPR v -> row v (+8 for high half)
    out[(size_t)m * HALF_ROWS + (size_t)(n0 + idx)] = c[v];
  }
}

// ---------------------------------------------------------------------------
extern "C" void kernel_launch(void* const* d_in, const int* in_sizes, int n_in,
                              void* d_out, int out_size, void* d_ws, size_t ws_size,
                              hipStream_t stream) {
  const float* core_first = (const float*)d_in[1];  // [1,2,32]
  const float* cores_mid  = (const float*)d_in[2];  // [20,32,2,32]
  const float* core_last  = (const float*)d_in[3];  // [32,2,1]
  float* out = (float*)d_out;                       // [2^22 state | 21 entropies]
  float* ws  = (float*)d_ws;

  float* La    = ws;                 // 65536 floats each
  float* Lb    = ws + 65536;
  float* Ra    = ws + 131072;
  float* Rb    = ws + 196608;
  float* grams = ws + 262144;        // 21 * 1024 floats

  // Entropy path (independent of state path)
  gram_chain_kernel<<<1, 1024, 0, stream>>>(core_first, cores_mid, grams);
  entropy_kernel<<<NBONDS, 32, 0, stream>>>(grams, out + STATE_SIZE);

  // Half-chain factors
  init_halves_kernel<<<1, 64, 0, stream>>>(core_first, core_last, La, Ra);

  float* lin = La; float* lout = Lb;
  int rows = 2;
  for (int k = 0; k < 10; ++k) {               // cores_mid[0..9]
    int total = rows * 2 * CHI;
    left_step_kernel<<<(total + 255) / 256, 256, 0, stream>>>(
        lin, cores_mid + k * 2048, lout, total);
    float* t = lin; lin = lout; lout = t;
    rows <<= 1;
  }
  float* Lfin = lin;                            // [2048,32]

  float* rin = Ra; float* rout = Rb;
  rows = 2; int rbits = 1;
  for (int k = 19; k >= 10; --k) {              // cores_mid[19..10]
    int total = rows * 2 * CHI;
    right_step_kernel<<<(total + 255) / 256, 256, 0, stream>>>(
        rin, cores_mid + k * 2048, rout, rbits, total);
    float* t = rin; rin = rout; rout = t;
    rows <<= 1; rbits++;
  }
  float* Rfin = rin;                            // [2048,32]

  dim3 grid(HALF_ROWS / 64, HALF_ROWS / 32);    // (32, 64)
  final_gemm_kernel<<<grid, 256, 0, stream>>>(Lfin, Rfin, out);
}